// GraphConvolutionalNetwork_47278999994618
// MI455X (gfx1250) — compile-verified
//
#include <hip/hip_runtime.h>
#include <hip/hip_bf16.h>
#include <math.h>

typedef float v2f __attribute__((ext_vector_type(2)));
typedef float v8f __attribute__((ext_vector_type(8)));

#if defined(__gfx1250__) && __has_builtin(__builtin_amdgcn_global_load_async_to_lds_b128)
#define GCN_ASYNC_LDS 1
typedef int gcn_v4i __attribute__((vector_size(16)));
typedef __attribute__((address_space(1))) gcn_v4i* gcn_g4p;
typedef __attribute__((address_space(3))) gcn_v4i* gcn_l4p;
#else
#define GCN_ASYNC_LDS 0
#endif

// ---------------------------------------------------------------------------
// Fused  out = act(A @ W^T + bias)  via V_WMMA_F32_16X16X4_F32.
// A: (N, K) row-major, W: (M, K) row-major, out: (N, M) row-major.
//
// Block = 160 threads (5 waves). Block computes an 80-row x 64-col output
// slab: the 64x K weight strip is staged ONCE into LDS (async-to-LDS when
// available) and shared by all 5 waves (5x cut in B-side L2 traffic). Each
// wave then computes a 16x64 strip: 4 f32 accumulators, A fragment loaded
// from global (software-pipelined, loop peeled so the prefetch is
// unconditional), B fragments via DS loads from LDS. LDS rows padded by
// +4 floats: keeps 16B store alignment and staggers rows across banks
// (K is a multiple of 64 -> unpadded rows would alias one bank).
//
// A-tile lane layout: row = lane&15, K = 2*(lane>>4) + {0,1}
// B-tile lane layout: col = lane&15, K = 2*(lane>>4) + {0,1}
// D layout: vgpr i, lane L -> row i + 8*(L>>4), col L&15.
// Requires: K % 4 == 0, M % 64 == 0, N % 80 == 0.
// ---------------------------------------------------------------------------
__global__ __launch_bounds__(160) void wmma_fc_kernel(
    const float* __restrict__ A, const float* __restrict__ W,
    const float* __restrict__ bias, float* __restrict__ out,
    int K, int Kp, int M, int do_relu)
{
    extern __shared__ float sW[];   // 64 * Kp floats

    const int tid  = threadIdx.x;
    const int wave = tid >> 5;
    const int lane = tid & 31;
    const int row0 = (blockIdx.x * 5 + wave) << 4;
    const int col0 = blockIdx.y << 6;

    // ---- Stage the 64xK weight strip (rows col0..col0+63 of W, contiguous)
    //      into LDS, float4 at a time, cooperatively across 160 threads. ----
    const float* __restrict__ wstrip = W + (size_t)col0 * K;
    const int K4 = K >> 2;
    const int total4 = 64 * K4;
    for (int i = tid; i < total4; i += 160) {
        const int rowi = i / K4;
        const int kk   = (i - rowi * K4) << 2;
        const float* gsrc = wstrip + (size_t)rowi * K + kk;
        float*       ldst = sW + (size_t)rowi * Kp + kk;
#if GCN_ASYNC_LDS
        __builtin_amdgcn_global_load_async_to_lds_b128(
            (gcn_g4p)(void*)gsrc, (gcn_l4p)(void*)ldst, 0, 0);
#else
        *(float4*)ldst = *(const float4*)gsrc;
#endif
    }
#if GCN_ASYNC_LDS
#if __has_builtin(__builtin_amdgcn_s_wait_asynccnt)
    __builtin_amdgcn_s_wait_asynccnt(0);
#else
    asm volatile("s_wait_asynccnt 0x0" ::: "memory");
#endif
#endif
    __syncthreads();

    // ---- WMMA main loop ----
    const int r  = lane & 15;            // A row / B col within 16-tile
    const int kb = (lane >> 4) << 1;     // 0 or 2

    const float* __restrict__ arow = A + (size_t)(row0 + r) * K + kb;
    const float* __restrict__ l0 = sW + (size_t)r * Kp + kb;
    const float* __restrict__ l1 = l0 + (size_t)16 * Kp;
    const float* __restrict__ l2 = l0 + (size_t)32 * Kp;
    const float* __restrict__ l3 = l0 + (size_t)48 * Kp;

    v8f acc0 = {}, acc1 = {}, acc2 = {}, acc3 = {};

    // Prologue: A fragment for k = 0.
    v2f ac;
    ac.x = arow[0]; ac.y = arow[1];

    for (int k = 4; k < K; k += 4) {
        // Unconditional prefetch of next A fragment (global, 1 step ahead).
        v2f an;
        an.x = arow[k]; an.y = arow[k + 1];

        const int kc = k - 4;            // current step's B fragments (LDS)
        v2f w0, w1, w2, w3;
        w0.x = l0[kc]; w0.y = l0[kc + 1];
        w1.x = l1[kc]; w1.y = l1[kc + 1];
        w2.x = l2[kc]; w2.y = l2[kc + 1];
        w3.x = l3[kc]; w3.y = l3[kc + 1];

        // (neg_a, A, neg_b, B, c_mod, C, reuse_a, reuse_b)
        acc0 = __builtin_amdgcn_wmma_f32_16x16x4_f32(false, ac, false, w0, (short)0, acc0, false, false);
        acc1 = __builtin_amdgcn_wmma_f32_16x16x4_f32(false, ac, false, w1, (short)0, acc1, false, false);
        acc2 = __builtin_amdgcn_wmma_f32_16x16x4_f32(false, ac, false, w2, (short)0, acc2, false, false);
        acc3 = __builtin_amdgcn_wmma_f32_16x16x4_f32(false, ac, false, w3, (short)0, acc3, false, false);

        ac = an;
    }

    // Epilogue: last k-step.
    {
        const int kc = K - 4;
        v2f w0, w1, w2, w3;
        w0.x = l0[kc]; w0.y = l0[kc + 1];
        w1.x = l1[kc]; w1.y = l1[kc + 1];
        w2.x = l2[kc]; w2.y = l2[kc + 1];
        w3.x = l3[kc]; w3.y = l3[kc + 1];
        acc0 = __builtin_amdgcn_wmma_f32_16x16x4_f32(false, ac, false, w0, (short)0, acc0, false, false);
        acc1 = __builtin_amdgcn_wmma_f32_16x16x4_f32(false, ac, false, w1, (short)0, acc1, false, false);
        acc2 = __builtin_amdgcn_wmma_f32_16x16x4_f32(false, ac, false, w2, (short)0, acc2, false, false);
        acc3 = __builtin_amdgcn_wmma_f32_16x16x4_f32(false, ac, false, w3, (short)0, acc3, false, false);
    }

    // ---- Bias + activation + store ----
    const int n     = lane & 15;
    const int rbase = (lane >> 4) << 3;
    v8f accs[4] = {acc0, acc1, acc2, acc3};
#pragma unroll
    for (int t = 0; t < 4; ++t) {
        const int col  = col0 + (t << 4) + n;
        const float bv = bias[col];
#pragma unroll
        for (int i = 0; i < 8; ++i) {
            float v = accs[t][i] + bv;
            if (do_relu) v = fmaxf(v, 0.0f);
            out[(size_t)(row0 + rbase + i) * M + col] = v;
        }
    }
}

// ---------------------------------------------------------------------------
__global__ void zero_kernel(float* __restrict__ p, int n) {
    int i = blockIdx.x * blockDim.x + threadIdx.x;
    if (i < n) p[i] = 0.0f;
}

__global__ void degree_kernel(const int* __restrict__ tgt, float* __restrict__ deg, int E) {
    int e = blockIdx.x * blockDim.x + threadIdx.x;
    if (e < E) atomicAdd(&deg[tgt[e]], 1.0f);
}

__global__ void inv_kernel(float* __restrict__ deg_inv, int n) {
    int i = blockIdx.x * blockDim.x + threadIdx.x;
    if (i < n) deg_inv[i] = 1.0f / (deg_inv[i] + 1.0f);
}

// One thread per (edge, 4-float chunk): float4 gather (global_load_b128 from
// L2-resident h) + 4 f32 atomic adds into agg (L2 atomic units).
__global__ void scatter_kernel(const float4* __restrict__ h4,
                               const int* __restrict__ src,
                               const int* __restrict__ tgt,
                               float* __restrict__ agg, int E, int C4)
{
    int idx = blockIdx.x * blockDim.x + threadIdx.x;
    if (idx >= E * C4) return;
    const int e = idx / C4;
    const int c = idx % C4;
    const float4 v = h4[(size_t)src[e] * C4 + c];
    float* dst = agg + ((size_t)tgt[e] * C4 + c) * 4;
    atomicAdd(dst + 0, v.x);
    atomicAdd(dst + 1, v.y);
    atomicAdd(dst + 2, v.z);
    atomicAdd(dst + 3, v.w);
}

__global__ void update_kernel(float* __restrict__ h, const float* __restrict__ agg,
                              const float* __restrict__ inv, int total, int C)
{
    int i = blockIdx.x * blockDim.x + threadIdx.x;
    if (i < total) {
        const int row = i / C;
        h[i] = 0.7f * h[i] + 0.3f * agg[i] * inv[row];
    }
}

// att[i] = sigmoid(dot(a1[i,:], Wa2) + ba2)
__global__ void att_kernel(const float* __restrict__ a1, const float* __restrict__ Wa2,
                           const float* __restrict__ ba2, float* __restrict__ att,
                           int N, int C)
{
    int i = blockIdx.x * blockDim.x + threadIdx.x;
    if (i >= N) return;
    float s = ba2[0];
    const float* row = a1 + (size_t)i * C;
    for (int c = 0; c < C; ++c) s += row[c] * Wa2[c];
    att[i] = 1.0f / (1.0f + expf(-s));
}

// ---------------------------------------------------------------------------
extern "C" void kernel_launch(void* const* d_in, const int* in_sizes, int n_in,
                              void* d_out, int out_size, void* d_ws, size_t ws_size,
                              hipStream_t stream)
{
    const float* x   = (const float*)d_in[0];
    const int*   ei  = (const int*)d_in[1];
    const float* W1  = (const float*)d_in[2];
    const float* b1  = (const float*)d_in[3];
    const float* W2  = (const float*)d_in[4];
    const float* b2  = (const float*)d_in[5];
    const float* Wa1 = (const float*)d_in[6];
    const float* ba1 = (const float*)d_in[7];
    const float* Wa2 = (const float*)d_in[8];
    const float* ba2 = (const float*)d_in[9];

    const int K1 = 384, H1 = 256, H2 = 128, HA = 64;
    const int N = in_sizes[0] / K1;   // 10000 (multiple of 80)
    const int E = in_sizes[1] / 2;    // 320000

    const int* src = ei;
    const int* tgt = ei + E;

    float* ws   = (float*)d_ws;
    float* h    = ws;                        // N*H1
    float* agg  = h   + (size_t)N * H1;      // N*H1
    float* inv  = agg + (size_t)N * H1;      // N (deg, then inverted in place)
    float* a1   = inv + N;                   // N*HA
    float* outp = (float*)d_out;             // N*H2
    float* att  = outp + (size_t)N * H2;     // N

    const int T = 256;

    // Allow ~100KB dynamic LDS for the fc1 weight strip (320KB/WGP on CDNA5).
    (void)hipFuncSetAttribute((const void*)wmma_fc_kernel,
                              hipFuncAttributeMaxDynamicSharedMemorySize,
                              64 * (K1 + 4) * (int)sizeof(float));

    // fc1: h = relu(x @ W1^T + b1)
    wmma_fc_kernel<<<dim3(N / 80, H1 / 64), 160, 64 * (K1 + 4) * sizeof(float), stream>>>(
        x, W1, b1, h, K1, K1 + 4, H1, 1);

    // degree -> inv = 1/(deg+1)
    zero_kernel<<<(N + T - 1) / T, T, 0, stream>>>(inv, N);
    degree_kernel<<<(E + T - 1) / T, T, 0, stream>>>(tgt, inv, E);
    inv_kernel<<<(N + T - 1) / T, T, 0, stream>>>(inv, N);

    // two propagation passes
    const int C4 = H1 / 4;
    for (int pass = 0; pass < 2; ++pass) {
        zero_kernel<<<((N * H1) + T - 1) / T, T, 0, stream>>>(agg, N * H1);
        scatter_kernel<<<((E * C4) + T - 1) / T, T, 0, stream>>>(
            (const float4*)h, src, tgt, agg, E, C4);
        update_kernel<<<((N * H1) + T - 1) / T, T, 0, stream>>>(h, agg, inv, N * H1, H1);
    }

    // fc2: out = relu(h @ W2^T + b2)
    wmma_fc_kernel<<<dim3(N / 80, H2 / 64), 160, 64 * (H1 + 4) * sizeof(float), stream>>>(
        h, W2, b2, outp, H1, H1 + 4, H2, 1);

    // attention layer 1: a1 = relu(out @ Wa1^T + ba1)
    wmma_fc_kernel<<<dim3(N / 80, HA / 64), 160, 64 * (H2 + 4) * sizeof(float), stream>>>(
        outp, Wa1, ba1, a1, H2, H2 + 4, HA, 1);

    // attention layer 2 + sigmoid
    att_kernel<<<(N + T - 1) / T, T, 0, stream>>>(a1, Wa2, ba2, att, N, HA);
}